// RelationalAttention_25778393711238
// MI455X (gfx1250) — compile-verified
//
#include <hip/hip_runtime.h>
#include <cmath>

typedef __attribute__((ext_vector_type(2))) float v2f;
typedef __attribute__((ext_vector_type(4))) float v4f;
typedef __attribute__((ext_vector_type(8))) float v8f;

#define BT     128      // B*T
#define NRES   64       // N
#define DMODEL 256      // D
#define NHEAD  4
#define HDIM   64
#define ROWS   (BT * NRES)   // 8192

// ---------------------------------------------------------------------------
// Kernel 1/4: tiled fp32 WMMA GEMM  C[ROWS, Ncols] = A[ROWS,256] * W[Ncols,256]^T + bias
// mode 0: scatter into head-major q/k/v  (Ncols = 768)
// mode 1: direct row-major store to out  (Ncols = 256), NT stores (host-consumed)
// Block: 256 threads (8 waves). Each block computes a 64x64 output tile;
// each wave computes two 16x16 tiles with V_WMMA_F32_16X16X4_F32.
// ---------------------------------------------------------------------------
__global__ __launch_bounds__(256) void gemm_wmma_f32(
    const float* __restrict__ A, const float* __restrict__ W,
    const float* __restrict__ bias,
    float* __restrict__ qd, float* __restrict__ kd, float* __restrict__ vd,
    float* __restrict__ outd, int mode)
{
    __shared__ float As[64 * 64];   // 16 KB: A rows x k-chunk
    __shared__ float Ws[64 * 64];   // 16 KB: W rows (out cols) x k-chunk

    const int tid  = threadIdx.x;
    const int lane = tid & 31;
    const int wid  = tid >> 5;
    const int row0 = blockIdx.y * 64;
    const int col0 = blockIdx.x * 64;

    const int t0 = wid * 2, t1 = t0 + 1;
    const int tm0 = t0 >> 2, tn0 = t0 & 3;
    const int tm1 = t1 >> 2, tn1 = t1 & 3;

    v8f acc0 = {}; v8f acc1 = {};

    const int l15  = lane & 15;
    const int koff = (lane >> 4) << 1;          // 0 for lanes 0-15, 2 for 16-31

    for (int k0 = 0; k0 < DMODEL; k0 += 64) {
        // Stage 64x64 chunks of A and W into LDS (b128 loads, fully coalesced)
#pragma unroll
        for (int i = 0; i < 4; ++i) {
            int idx = tid + i * 256;            // 1024 float4s per array
            int r = idx >> 4;
            int c = (idx & 15) << 2;
            *(float4*)&As[r * 64 + c] = *(const float4*)&A[(size_t)(row0 + r) * DMODEL + k0 + c];
            *(float4*)&Ws[r * 64 + c] = *(const float4*)&W[(size_t)(col0 + r) * DMODEL + k0 + c];
        }
        __syncthreads();

        const int am0 = (tm0 * 16 + l15) * 64;
        const int am1 = (tm1 * 16 + l15) * 64;
        const int bn0 = (tn0 * 16 + l15) * 64;
        const int bn1 = (tn1 * 16 + l15) * 64;
#pragma unroll
        for (int kk = 0; kk < 64; kk += 4) {
            const int kb = kk + koff;
            v2f a0 = *(const v2f*)&As[am0 + kb];
            v2f b0 = *(const v2f*)&Ws[bn0 + kb];
            acc0 = __builtin_amdgcn_wmma_f32_16x16x4_f32(false, a0, false, b0,
                                                         (short)0, acc0, false, false);
            v2f a1 = *(const v2f*)&As[am1 + kb];
            v2f b1 = *(const v2f*)&Ws[bn1 + kb];
            acc1 = __builtin_amdgcn_wmma_f32_16x16x4_f32(false, a1, false, b1,
                                                         (short)0, acc1, false, false);
        }
        __syncthreads();
    }

    // Epilogue: C/D layout — VGPR r: lanes 0-15 -> M=r, lanes 16-31 -> M=r+8
#pragma unroll
    for (int i = 0; i < 2; ++i) {
        const v8f acc = i ? acc1 : acc0;
        const int tm = i ? tm1 : tm0;
        const int tn = i ? tn1 : tn0;
        const int gc = col0 + tn * 16 + l15;
        const float bv = bias[gc];
#pragma unroll
        for (int r = 0; r < 8; ++r) {
            const int gr = row0 + tm * 16 + r + ((lane >> 4) << 3);
            const float val = acc[r] + bv;
            if (mode == 1) {
                // final output: host-consumed, keep it out of the hot L2 set
                __builtin_nontemporal_store(val, &outd[(size_t)gr * DMODEL + gc]);
            } else {
                const int m = gr >> 6, n = gr & 63;
                const int part = gc >> 8, c = gc & 255;
                const int h = c >> 6, hd = c & 63;
                float* dst = (part == 0) ? qd : ((part == 1) ? kd : vd);
                dst[(((size_t)m * NHEAD + h) * NRES + n) * HDIM + hd] = val;
            }
        }
    }
}

// ---------------------------------------------------------------------------
// Kernel 2: bias projection  bmat[m,h,n,p] = sum_d bias_features[m,n,p,d]*Wbias[h,d]
// Pure HBM stream of 512 MB (>> 192 MB L2, touched exactly once): use
// NON-TEMPORAL b128 loads so the stream does not evict the q/k/v/bmat working
// set that the attention kernel re-reads from L2.
// One wave per row of 256 floats, 4 head dot-products, wave32 xor-shuffle
// reduce, lanes 0..3 store (bmat stays RT — consumed by kernel 3 from L2).
// ---------------------------------------------------------------------------
__global__ __launch_bounds__(256) void bias_proj_kernel(
    const float* __restrict__ bf,      // [Bt*N*N, 256]
    const float* __restrict__ Wbias,   // [4, 256]
    float* __restrict__ bmat)          // [Bt, H, N, N]
{
    const int lane = threadIdx.x & 31;
    const int wid  = threadIdx.x >> 5;
    const long waveId = (long)blockIdx.x * 8 + wid;

    v4f w0[4], w1[4];
#pragma unroll
    for (int h = 0; h < 4; ++h) {
        w0[h] = *(const v4f*)&Wbias[h * DMODEL + lane * 8];
        w1[h] = *(const v4f*)&Wbias[h * DMODEL + lane * 8 + 4];
    }

    for (int rr = 0; rr < 8; ++rr) {
        const long R = waveId * 8 + rr;                  // row in [0, 524288)
        const float* p = bf + (size_t)R * DMODEL + lane * 8;
        __builtin_prefetch(p + DMODEL, 0, 0);            // global_prefetch_b8 next row
        const v4f x0 = __builtin_nontemporal_load((const v4f*)p);       // th:NT
        const v4f x1 = __builtin_nontemporal_load((const v4f*)(p + 4)); // th:NT

        float s[4];
#pragma unroll
        for (int h = 0; h < 4; ++h) {
            s[h] = x0.x * w0[h].x + x0.y * w0[h].y + x0.z * w0[h].z + x0.w * w0[h].w
                 + x1.x * w1[h].x + x1.y * w1[h].y + x1.z * w1[h].z + x1.w * w1[h].w;
        }
#pragma unroll
        for (int h = 0; h < 4; ++h) {
#pragma unroll
            for (int off = 16; off >= 1; off >>= 1)
                s[h] += __shfl_xor(s[h], off, 32);
        }
        const int m = (int)(R >> 12);
        const int n = (int)((R >> 6) & 63);
        const int pc = (int)(R & 63);
        if (lane < 4)
            bmat[((((size_t)m * NHEAD) + lane) * NRES + n) * NRES + pc] = s[lane];
    }
}

// ---------------------------------------------------------------------------
// Kernel 3: attention per (m,h).  LDS-resident 64x64 Q/K/V tiles + score tile.
// scores = (Q K^T)/8 + bmat (masked), softmax rows, out = A V  — WMMA both GEMMs.
// ---------------------------------------------------------------------------
__global__ __launch_bounds__(256) void attn_kernel(
    const float* __restrict__ q, const float* __restrict__ k,
    const float* __restrict__ v, const float* __restrict__ bmat,
    const unsigned char* __restrict__ mask,   // [Bt*N] bool
    float* __restrict__ aout)                 // [Bt*N, 256]
{
    __shared__ float qs[64 * 64];
    __shared__ float ks[64 * 64];
    __shared__ float vs[64 * 64];
    __shared__ float sc[64 * 64];

    const int mh = blockIdx.x;          // m*H + h
    const int m  = mh >> 2, h = mh & 3;
    const int tid = threadIdx.x, lane = tid & 31, wid = tid >> 5;
    const int l15 = lane & 15;
    const int koff = (lane >> 4) << 1;

    const float* qg = q + (size_t)mh * 64 * 64;
    const float* kg = k + (size_t)mh * 64 * 64;
    const float* vg = v + (size_t)mh * 64 * 64;
    const float* bg = bmat + (size_t)mh * 64 * 64;

#pragma unroll
    for (int i = 0; i < 4; ++i) {
        int idx = (tid + i * 256) << 2;
        *(float4*)&qs[idx] = *(const float4*)&qg[idx];
        *(float4*)&ks[idx] = *(const float4*)&kg[idx];
        *(float4*)&vs[idx] = *(const float4*)&vg[idx];
    }
    __syncthreads();

    const int t0 = wid * 2, t1 = t0 + 1;
    const int tm0 = t0 >> 2, tn0 = t0 & 3;
    const int tm1 = t1 >> 2, tn1 = t1 & 3;

    // ---- scores = Q K^T ----
    {
        v8f s0 = {}; v8f s1 = {};
        const int am0 = (tm0 * 16 + l15) * 64;
        const int am1 = (tm1 * 16 + l15) * 64;
        const int bn0 = (tn0 * 16 + l15) * 64;   // B = K^T, row-major K works directly
        const int bn1 = (tn1 * 16 + l15) * 64;
#pragma unroll
        for (int kk = 0; kk < 64; kk += 4) {
            const int kb = kk + koff;
            v2f a0 = *(const v2f*)&qs[am0 + kb];
            v2f b0 = *(const v2f*)&ks[bn0 + kb];
            s0 = __builtin_amdgcn_wmma_f32_16x16x4_f32(false, a0, false, b0,
                                                       (short)0, s0, false, false);
            v2f a1 = *(const v2f*)&qs[am1 + kb];
            v2f b1 = *(const v2f*)&ks[bn1 + kb];
            s1 = __builtin_amdgcn_wmma_f32_16x16x4_f32(false, a1, false, b1,
                                                       (short)0, s1, false, false);
        }
        // scale + relational bias + mask -> LDS score tile
#pragma unroll
        for (int i = 0; i < 2; ++i) {
            const v8f s = i ? s1 : s0;
            const int tm = i ? tm1 : tm0;
            const int tn = i ? tn1 : tn0;
            const int col = tn * 16 + l15;
            const bool keep = mask[m * NRES + col] != 0;
#pragma unroll
            for (int r = 0; r < 8; ++r) {
                const int row = tm * 16 + r + ((lane >> 4) << 3);
                float val = s[r] * 0.125f + bg[row * 64 + col];
                val = keep ? val : -3.402823466e+38f;
                sc[row * 64 + col] = val;
            }
        }
    }
    __syncthreads();

    // ---- softmax over rows: 4 threads per row, 16 cols each ----
    {
        const int row = tid >> 2;
        const int seg = (tid & 3) << 4;
        float* rp = &sc[row * 64 + seg];
        float mx = -3.402823466e+38f;
#pragma unroll
        for (int j = 0; j < 16; ++j) mx = fmaxf(mx, rp[j]);
        mx = fmaxf(mx, __shfl_xor(mx, 1, 4));
        mx = fmaxf(mx, __shfl_xor(mx, 2, 4));
        float sum = 0.0f;
#pragma unroll
        for (int j = 0; j < 16; ++j) { float e = __expf(rp[j] - mx); rp[j] = e; sum += e; }
        sum += __shfl_xor(sum, 1, 4);
        sum += __shfl_xor(sum, 2, 4);
        const float inv = 1.0f / sum;
#pragma unroll
        for (int j = 0; j < 16; ++j) rp[j] *= inv;
    }
    __syncthreads();

    // ---- out = A V ----
    {
        v8f o0 = {}; v8f o1 = {};
        const int am0 = (tm0 * 16 + l15) * 64;
        const int am1 = (tm1 * 16 + l15) * 64;
        const int n0  = tn0 * 16 + l15;
        const int n1  = tn1 * 16 + l15;
#pragma unroll
        for (int kk = 0; kk < 64; kk += 4) {
            const int kb = kk + koff;
            v2f a0 = *(const v2f*)&sc[am0 + kb];
            v2f b0; b0.x = vs[kb * 64 + n0]; b0.y = vs[(kb + 1) * 64 + n0];
            o0 = __builtin_amdgcn_wmma_f32_16x16x4_f32(false, a0, false, b0,
                                                       (short)0, o0, false, false);
            v2f a1 = *(const v2f*)&sc[am1 + kb];
            v2f b1; b1.x = vs[kb * 64 + n1]; b1.y = vs[(kb + 1) * 64 + n1];
            o1 = __builtin_amdgcn_wmma_f32_16x16x4_f32(false, a1, false, b1,
                                                       (short)0, o1, false, false);
        }
#pragma unroll
        for (int i = 0; i < 2; ++i) {
            const v8f o = i ? o1 : o0;
            const int tm = i ? tm1 : tm0;
            const int tn = i ? tn1 : tn0;
            const int col = tn * 16 + l15;
#pragma unroll
            for (int r = 0; r < 8; ++r) {
                const int row = tm * 16 + r + ((lane >> 4) << 3);
                aout[((size_t)m * NRES + row) * DMODEL + h * HDIM + col] = o[r];
            }
        }
    }
}

// ---------------------------------------------------------------------------
extern "C" void kernel_launch(void* const* d_in, const int* in_sizes, int n_in,
                              void* d_out, int out_size, void* d_ws, size_t ws_size,
                              hipStream_t stream) {
    const float* x     = (const float*)d_in[0];
    const float* bf    = (const float*)d_in[1];
    const unsigned char* mask = (const unsigned char*)d_in[2];
    const float* Wqkv  = (const float*)d_in[3];
    const float* bqkv  = (const float*)d_in[4];
    const float* Wproj = (const float*)d_in[5];
    const float* bproj = (const float*)d_in[6];
    const float* Wbias = (const float*)d_in[7];
    float* out = (float*)d_out;

    float* ws = (float*)d_ws;
    const size_t SEG = (size_t)BT * NHEAD * NRES * HDIM;   // 2,097,152 floats
    float* qd = ws;
    float* kd = qd + SEG;
    float* vd = kd + SEG;
    float* bm = vd + SEG;          // [Bt,H,N,N] same element count
    float* ao = bm + SEG;          // [Bt*N, 256]

    dim3 blk(256);
    // 1) QKV projection: [8192,256] x [256,768]
    gemm_wmma_f32<<<dim3(768 / 64, ROWS / 64), blk, 0, stream>>>(
        x, Wqkv, bqkv, qd, kd, vd, nullptr, 0);
    // 2) Relational bias projection: streams 512 MB once (NT loads)
    bias_proj_kernel<<<(BT * NRES * NRES) / 64, blk, 0, stream>>>(bf, Wbias, bm);
    // 3) Attention per (m,h)
    attn_kernel<<<BT * NHEAD, blk, 0, stream>>>(qd, kd, vd, bm, mask, ao);
    // 4) Output projection: [8192,256] x [256,256]
    gemm_wmma_f32<<<dim3(DMODEL / 64, ROWS / 64), blk, 0, stream>>>(
        ao, Wproj, bproj, nullptr, nullptr, nullptr, out, 1);
}